// Dual_Quantize2_43645457662415
// MI455X (gfx1250) — compile-verified
//
#include <hip/hip_runtime.h>
#include <hip/hip_bf16.h>

// ---------------------------------------------------------------------------
// Dual vector-quantization (parallel forward, eval) for MI455X / gfx1250.
//   dist = ||f||^2 - 2 f.E + ||E||^2 ; argmin over 1024 codes; gather; MSE.
// Core matmul: V_WMMA_F32_16X16X4_F32 (wave32 WMMA).
// LDS staging:  GLOBAL_LOAD_ASYNC_TO_LDS_B128/B32 (ASYNCcnt) -- no VGPR trip.
// ---------------------------------------------------------------------------

#define DIMQ   256
#define NE     1024
#define KTOT   512
#define NROWS  65536

typedef __attribute__((ext_vector_type(2))) float v2f;
typedef __attribute__((ext_vector_type(8))) float v8f;

// ---------------- output layout (floats, concatenated in return order) -----
#define OUT_QHR  ((size_t)0)
#define OUT_QLR  ((size_t)16777216)
#define OUT_DHR  ((size_t)33554432)
#define OUT_DLR  ((size_t)33554433)
#define OUT_IND0 ((size_t)33554434)
#define OUT_IND1 ((size_t)33619970)

// ---------------- workspace layout (float units) ---------------------------
#define WS_NORME   ((size_t)0)          // 1024 f32
#define WS_IND     ((size_t)1024)       // 65536 i32
#define WS_TLR     ((size_t)66560)      // 1024x256 f32 (embed_lr^T)
#define WS_THR     ((size_t)328704)     // 1024x256 f32 (embed_hr^T)
#define WS_PHR     ((size_t)590848)     // 16384 f32 partial sums
#define WS_PLR     ((size_t)607232)     // 16384 f32 partial sums

// ---------------- CDNA5 async global->LDS helpers --------------------------
__device__ __forceinline__ void async_load_b128(unsigned lds_off, const void* g) {
  asm volatile("global_load_async_to_lds_b128 %0, %1, off"
               :: "v"(lds_off), "v"((unsigned long long)(uintptr_t)g) : "memory");
}
__device__ __forceinline__ void async_load_b32(unsigned lds_off, const void* g) {
  asm volatile("global_load_async_to_lds_b32 %0, %1, off"
               :: "v"(lds_off), "v"((unsigned long long)(uintptr_t)g) : "memory");
}
__device__ __forceinline__ void async_wait0() {
#if defined(__has_builtin)
#if __has_builtin(__builtin_amdgcn_s_wait_asynccnt)
  __builtin_amdgcn_s_wait_asynccnt(0);
#else
  asm volatile("s_wait_asynccnt 0x0" ::: "memory");
#endif
#else
  asm volatile("s_wait_asynccnt 0x0" ::: "memory");
#endif
}

// ===========================================================================
// Kernel 1: transpose both codebooks (dim, n_embed) -> (n_embed, dim)
// ===========================================================================
__global__ void k_transpose(const float* __restrict__ emb_lr,
                            const float* __restrict__ emb_hr,
                            float* __restrict__ tlr,
                            float* __restrict__ thr) {
  __shared__ float tile[32][33];
  const float* src = blockIdx.z ? emb_hr : emb_lr;
  float*       dst = blockIdx.z ? thr    : tlr;
  const int jb = blockIdx.x * 32;           // column block in [0,1024)
  const int db = blockIdx.y * 32;           // row block in [0,256)
  const int tx = threadIdx.x & 31;
  const int ty = threadIdx.x >> 5;          // 32x8 threads
#pragma unroll
  for (int i = 0; i < 32; i += 8)
    tile[ty + i][tx] = src[(size_t)(db + ty + i) * NE + jb + tx];
  __syncthreads();
#pragma unroll
  for (int i = 0; i < 32; i += 8)
    dst[(size_t)(jb + ty + i) * DIMQ + db + tx] = tile[tx][ty + i];
}

// ===========================================================================
// Kernel 2: ||E_j||^2 over the concatenated (lr,hr) codebook columns
// ===========================================================================
__global__ void k_norm(const float* __restrict__ tlr,
                       const float* __restrict__ thr,
                       float* __restrict__ normE) {
  const int j = blockIdx.x * blockDim.x + threadIdx.x;
  if (j >= NE) return;
  const float4* a = (const float4*)(tlr + (size_t)j * DIMQ);
  const float4* b = (const float4*)(thr + (size_t)j * DIMQ);
  float s = 0.f;
#pragma unroll 4
  for (int i = 0; i < DIMQ / 4; ++i) {
    float4 x = a[i];
    float4 y = b[i];
    s += x.x * x.x + x.y * x.y + x.z * x.z + x.w * x.w;
    s += y.x * y.x + y.y * y.y + y.z * y.z + y.w * y.w;
  }
  normE[j] = s;
}

// ===========================================================================
// Kernel 3: distances via V_WMMA_F32_16X16X4_F32 + fused argmin.
//   Grid: 512 WGs x 256 threads (8 wave32). WG owns 128 rows; each wave owns
//   16 rows and all 1024 columns (8 col-blocks of 128 -> 8 C tiles in regs).
//   Panels staged into LDS with async global->LDS DMA (ASYNCcnt).
// ===========================================================================
__global__ __launch_bounds__(256)
void k_dist(const float* __restrict__ in_hr,
            const float* __restrict__ in_lr,
            const float* __restrict__ emb_lr,
            const float* __restrict__ emb_hr,
            const float* __restrict__ normE,
            int* __restrict__ ind) {
  __shared__ float As[128][36];   // [row][k]  32-wide K panel (+4 pad, 16B rows)
  __shared__ float Bs[128][36];   // [col][k]  transposed B panel

  const int tid     = threadIdx.x;
  const int lane    = tid & 31;
  const int wave    = tid >> 5;
  const int rowbase = blockIdx.x * 128;
  const int aRow    = wave * 16 + (lane & 15);   // WMMA A: M = lane%16
  const int hi2     = (lane >> 4) * 2;           // lanes 16..31 carry K+2,K+3

  float minv[8];
  int   mini[8];
#pragma unroll
  for (int r = 0; r < 8; ++r) { minv[r] = 3.4e38f; mini[r] = 0; }

  for (int cb = 0; cb < 8; ++cb) {
    const int colb = cb * 128;
    v8f acc[8];
#pragma unroll
    for (int t = 0; t < 8; ++t) acc[t] = (v8f)(0.0f);

    for (int p = 0; p < 16; ++p) {
      const int k0 = p * 32;                   // panel fully in lr or hr half
      const int kb = k0 & 255;
      __syncthreads();                         // previous panel fully consumed
      // ---- stage A panel: 128 rows x 32 k, async b128 (4 per thread) ----
      {
        const float* src = (k0 < 256) ? in_lr : in_hr;
#pragma unroll
        for (int i = 0; i < 4; ++i) {
          const int idx = i * 256 + tid;       // float4 index in panel
          const int r = idx >> 3, k4 = (idx & 7) * 4;
          async_load_b128((unsigned)(uintptr_t)&As[r][k4],
                          src + (size_t)(rowbase + r) * DIMQ + kb + k4);
        }
      }
      // ---- stage B panel: 32 k x 128 cols, async b32 transpose-scatter ----
      {
        const float* srcE = (k0 < 256) ? emb_lr : emb_hr;
#pragma unroll
        for (int i = 0; i < 16; ++i) {
          const int idx = i * 256 + tid;
          const int kk = idx >> 7, c = idx & 127;
          async_load_b32((unsigned)(uintptr_t)&Bs[c][kk],
                         srcE + (size_t)(kb + kk) * NE + colb + c);
        }
      }
      async_wait0();                           // my async writes landed in LDS
      __syncthreads();                         // everyone's writes visible
      // ---- 8 k-steps x 8 column tiles of f32 WMMA ----
      for (int kk = 0; kk < 32; kk += 4) {
        const v2f a = *(const v2f*)&As[aRow][kk + hi2];
#pragma unroll
        for (int t = 0; t < 8; ++t) {
          const v2f b = *(const v2f*)&Bs[t * 16 + (lane & 15)][kk + hi2];
          acc[t] = __builtin_amdgcn_wmma_f32_16x16x4_f32(
              /*neg_a=*/false, a, /*neg_b=*/false, b,
              /*c_mod=*/(short)0, acc[t], /*reuse_a=*/false, /*reuse_b=*/false);
        }
      }
    }
    // ---- fused argmin update: dist = ||E||^2 - 2*dot (||f||^2 is const) ----
#pragma unroll
    for (int t = 0; t < 8; ++t) {
      const int col = colb + t * 16 + (lane & 15);
      const float ne = normE[col];
#pragma unroll
      for (int r = 0; r < 8; ++r) {
        const float d = ne - 2.0f * acc[t][r];
        if (d < minv[r]) { minv[r] = d; mini[r] = col; }  // cols increase -> first-min kept
      }
    }
  }
  // ---- reduce (min, idx) across the 16-lane groups, tie-break on low idx ----
#pragma unroll
  for (int r = 0; r < 8; ++r) {
    float v = minv[r];
    int  ix = mini[r];
#pragma unroll
    for (int m = 1; m < 16; m <<= 1) {
      const float ov = __shfl_xor(v, m, 32);
      const int   oi = __shfl_xor(ix, m, 32);
      if (ov < v || (ov == v && oi < ix)) { v = ov; ix = oi; }
    }
    if ((lane & 15) == 0) {
      const int row = rowbase + wave * 16 + r + ((lane >> 4) << 3); // hi lanes = M+8
      ind[row] = ix;
    }
  }
}

// ===========================================================================
// Kernel 4: coalesced gather from transposed codebooks + MSE partials + inds
// ===========================================================================
__global__ void k_gather(const float* __restrict__ in_hr,
                         const float* __restrict__ in_lr,
                         const float* __restrict__ tlr,
                         const float* __restrict__ thr,
                         const int* __restrict__ ind,
                         float* __restrict__ out,
                         float* __restrict__ part_hr,
                         float* __restrict__ part_lr) {
  const int b   = blockIdx.x;     // 16384 blocks x 4 rows
  const int tid = threadIdx.x;    // 256 = DIMQ
  float shr = 0.f, slr = 0.f;
#pragma unroll
  for (int rr = 0; rr < 4; ++rr) {
    const int n = b * 4 + rr;
    const int j = ind[n];
    const float qh = thr[(size_t)j * DIMQ + tid];
    const float ql = tlr[(size_t)j * DIMQ + tid];
    const float ih = in_hr[(size_t)n * DIMQ + tid];
    const float il = in_lr[(size_t)n * DIMQ + tid];
    out[OUT_QHR + (size_t)n * DIMQ + tid] = qh;
    out[OUT_QLR + (size_t)n * DIMQ + tid] = ql;
    const float dh = qh - ih, dl = ql - il;
    shr += dh * dh;
    slr += dl * dl;
  }
  __shared__ float red[2][256];
  red[0][tid] = shr;
  red[1][tid] = slr;
  __syncthreads();
  for (int s = 128; s > 0; s >>= 1) {
    if (tid < s) {
      red[0][tid] += red[0][tid + s];
      red[1][tid] += red[1][tid + s];
    }
    __syncthreads();
  }
  if (tid == 0) { part_hr[b] = red[0][0]; part_lr[b] = red[1][0]; }
  if (tid < 4) {
    const int n = b * 4 + tid;
    const float fi = (float)ind[n];
    out[OUT_IND0 + n] = fi;
    out[OUT_IND1 + n] = fi;
  }
}

// ===========================================================================
// Kernel 5: final deterministic reduction of the MSE partials
// ===========================================================================
__global__ void k_final(const float* __restrict__ part_hr,
                        const float* __restrict__ part_lr,
                        float* __restrict__ out) {
  __shared__ float red[2][256];
  const int tid = threadIdx.x;
  float sh = 0.f, sl = 0.f;
  for (int i = tid; i < 16384; i += 256) { sh += part_hr[i]; sl += part_lr[i]; }
  red[0][tid] = sh;
  red[1][tid] = sl;
  __syncthreads();
  for (int s = 128; s > 0; s >>= 1) {
    if (tid < s) {
      red[0][tid] += red[0][tid + s];
      red[1][tid] += red[1][tid + s];
    }
    __syncthreads();
  }
  if (tid == 0) {
    const float inv = 1.0f / ((float)NROWS * (float)DIMQ);
    out[OUT_DHR] = red[0][0] * inv;
    out[OUT_DLR] = red[1][0] * inv;
  }
}

// ===========================================================================
extern "C" void kernel_launch(void* const* d_in, const int* in_sizes, int n_in,
                              void* d_out, int out_size, void* d_ws, size_t ws_size,
                              hipStream_t stream) {
  const float* in_hr  = (const float*)d_in[0];
  const float* in_lr  = (const float*)d_in[1];
  const float* emb_lr = (const float*)d_in[2];
  const float* emb_hr = (const float*)d_in[3];
  float* out = (float*)d_out;
  float* ws  = (float*)d_ws;

  float* normE   = ws + WS_NORME;
  int*   ind     = (int*)(ws + WS_IND);
  float* tlr     = ws + WS_TLR;
  float* thr     = ws + WS_THR;
  float* part_hr = ws + WS_PHR;
  float* part_lr = ws + WS_PLR;

  k_transpose<<<dim3(32, 8, 2), 256, 0, stream>>>(emb_lr, emb_hr, tlr, thr);
  k_norm<<<4, 256, 0, stream>>>(tlr, thr, normE);
  k_dist<<<NROWS / 128, 256, 0, stream>>>(in_hr, in_lr, emb_lr, emb_hr, normE, ind);
  k_gather<<<NROWS / 4, 256, 0, stream>>>(in_hr, in_lr, tlr, thr, ind, out, part_hr, part_lr);
  k_final<<<1, 256, 0, stream>>>(part_hr, part_lr, out);
}